// LevenshteinDistance_72636486910443
// MI455X (gfx1250) — compile-verified
//
#include <hip/hip_runtime.h>

// Problem geometry (structural: 32 lanes x 16 columns == L)
#define LSEQ 512
#define KCOL 16
#define WAVES_PER_BLK 8

// ---------------------------------------------------------------------------
// Levenshtein wavefront DP: one wave32 per (source,query) pair.
// Lane t owns columns [16t+1 .. 16t+16]. Skewed schedule: at step s lane t
// computes DP row i = s - t + 1 of its stripe (registers only). Left/diag
// boundary values flow lane t-1 -> lane t via __shfl_up each step.
// Source row staged to LDS with CDNA5 async global->LDS loads (ASYNCcnt).
// ---------------------------------------------------------------------------
__global__ __launch_bounds__(256)
void lev_wavefront_kernel(const int* __restrict__ sources,
                          const int* __restrict__ queries,
                          float* __restrict__ dist,
                          int C, int P) {
  __shared__ int lds_src[WAVES_PER_BLK][LSEQ];

  const int lane = (int)(threadIdx.x & 31u);
  const int w    = (int)(threadIdx.x >> 5);
  const int p    = (int)blockIdx.x * WAVES_PER_BLK + w;
  if (p >= P) return;  // uniform per wave; no cross-wave barriers used

  const int* __restrict__ src = sources + (size_t)p * LSEQ;           // DP rows
  const int* __restrict__ qrw = queries + (size_t)(p / C) * LSEQ;     // DP cols

  // ---- stage source row -> LDS via async copy engine (gfx1250) ----
#if defined(__gfx1250__)
  {
    // generic LDS pointer: low 32 bits are the LDS byte offset
    unsigned ldsBase = (unsigned)(unsigned long long)&lds_src[w][0];
    #pragma unroll
    for (int r = 0; r < 4; ++r) {
      unsigned long long ga = (unsigned long long)(src + r * 128 + lane * 4);
      unsigned la = ldsBase + (unsigned)((r * 128 + lane * 4) * 4);
      // per-lane 16B: 32 lanes x 16B = 512B per instr, 4 instrs = 2KB row
      asm volatile("global_load_async_to_lds_b128 %0, %1, off"
                   :: "v"(la), "v"(ga) : "memory");
    }
  }
#else
  for (int idx = lane; idx < LSEQ; idx += 32) lds_src[w][idx] = src[idx];
#endif

  // ---- query stripe into registers (coalesced 16B loads) ----
  int btok[KCOL];
  {
    const int4* q4 = reinterpret_cast<const int4*>(qrw + lane * KCOL);
    #pragma unroll
    for (int r = 0; r < KCOL / 4; ++r) {
      int4 v = q4[r];
      btok[r * 4 + 0] = v.x; btok[r * 4 + 1] = v.y;
      btok[r * 4 + 2] = v.z; btok[r * 4 + 3] = v.w;
    }
  }

  // ---- DP state: prev[] = D[i-1][cols], boundary regs lb0/lb1 ----
  int prev[KCOL];
  #pragma unroll
  for (int k = 0; k < KCOL; ++k) prev[k] = lane * KCOL + k + 1;  // D[0][j] = j
  int lb1 = (lane == 0) ? 1 : lane * KCOL;   // left  boundary D[i][16t]
  int lb0 = (lane == 0) ? 0 : lane * KCOL;   // diag  boundary D[i-1][16t]
  int lastNew = 0;

#if defined(__gfx1250__)
  asm volatile("s_wait_asynccnt 0" ::: "memory");
#endif

  const int* srow = &lds_src[w][0];
  for (int s = 0; s < LSEQ + 31; ++s) {
    const int ri = s - lane;                  // row-1 this lane computes
    if (ri >= 0 && ri < LSEQ) {
      const int ai = srow[ri];                // conflict-free: 4B stride/lane
      int left = lb1, diag = lb0;
      #pragma unroll
      for (int k = 0; k < KCOL; ++k) {
        int up = prev[k];
        int c  = diag + (ai != btok[k] ? 1 : 0);
        int ul = (up < left ? up : left) + 1;
        int m  = (c < ul ? c : ul);
        diag   = up;
        prev[k] = m;
        left   = m;
      }
      lastNew = left;                         // D[i][16t+16]
    }
    // pass last column up one lane (wave32 lockstep, no barrier)
    int recv = __shfl_up(lastNew, 1);
    if (lane == 0) recv = s + 2;              // D[i+1][0] = i+1
    if (s >= lane - 1) { lb0 = lb1; lb1 = recv; }
  }

  if (lane == 31) dist[p] = (float)prev[KCOL - 1] + 0.0001f;  // D[512][512]
}

// ---------------------------------------------------------------------------
// Per query: first-max argmax over C distances (matches jnp.argmax ties),
// copy winning source row as float, write 10-wide distance table.
// ---------------------------------------------------------------------------
__global__ __launch_bounds__(256)
void lev_select_kernel(const int* __restrict__ sources,
                       const float* __restrict__ dist,
                       float* __restrict__ out,
                       int B, int C) {
  const int b = (int)blockIdx.x;
  float best = -1.0f; int bi = 0;
  for (int c = 0; c < C; ++c) {
    float v = dist[b * C + c];
    if (v > best) { best = v; bi = c; }       // strict > => first max
  }
  const int* src = sources + (size_t)(b * C + bi) * LSEQ;
  for (int t = (int)threadIdx.x; t < LSEQ; t += (int)blockDim.x)
    out[(size_t)b * LSEQ + t] = (float)src[t];
  if ((int)threadIdx.x < 10) {
    int c = (int)threadIdx.x;
    out[(size_t)B * LSEQ + b * 10 + c] = (c < C) ? dist[b * C + c] : 0.0f;
  }
}

extern "C" void kernel_launch(void* const* d_in, const int* in_sizes, int n_in,
                              void* d_out, int out_size, void* d_ws, size_t ws_size,
                              hipStream_t stream) {
  const int* sources = (const int*)d_in[0];   // [B*C, 512]
  const int* queries = (const int*)d_in[1];   // [B, 512]
  const int C = in_sizes[0] / in_sizes[1];    // = 10
  const int B = in_sizes[1] / LSEQ;           // = 512
  const int P = B * C;                        // = 5120 pairs
  float* dist = (float*)d_ws;                 // P floats of scratch

  const int blocks = (P + WAVES_PER_BLK - 1) / WAVES_PER_BLK;
  hipLaunchKernelGGL(lev_wavefront_kernel, dim3(blocks), dim3(256), 0, stream,
                     sources, queries, dist, C, P);
  hipLaunchKernelGGL(lev_select_kernel, dim3(B), dim3(256), 0, stream,
                     sources, dist, (float*)d_out, B, C);
}